// HebbianEncoder_16183436772192
// MI455X (gfx1250) — compile-verified
//
#include <hip/hip_runtime.h>
#include <math.h>

typedef __attribute__((ext_vector_type(2))) float v2f;
typedef __attribute__((ext_vector_type(8))) float v8f;

#define B_   32
#define CIN_ 3
#define HIN_ 64
#define WIN_ 64
#define C_   64
#define N_   1024          // 32*32 flattened spatial
#define R_   (B_ * C_)     // 2048 rows of flattened Af

// ---------------------------------------------------------------------------
// Kernel 1: conv3x3 s2 p1 + bias + ReLU, per-(b,c) Frobenius normalize -> af
// One block per (b,c); 256 threads, 4 output pixels each.
// ---------------------------------------------------------------------------
__global__ void conv_norm_kernel(const float* __restrict__ x,
                                 const float* __restrict__ w,
                                 const float* __restrict__ bias,
                                 float* __restrict__ af) {
  int bc = blockIdx.x;
  int b = bc / C_, c = bc % C_;
  __shared__ float sw[CIN_ * 9];
  __shared__ float sred[256];
  int tid = threadIdx.x;
  if (tid < CIN_ * 9) sw[tid] = w[c * CIN_ * 9 + tid];
  __syncthreads();
  float bval = bias[c];
  float vals[4];
  float ss = 0.f;
#pragma unroll
  for (int p = 0; p < 4; ++p) {
    int n = tid + p * 256;
    int oh = n >> 5, ow = n & 31;
    float acc = bval;
#pragma unroll
    for (int ci = 0; ci < CIN_; ++ci) {
      const float* xp = x + ((size_t)(b * CIN_ + ci) * HIN_) * WIN_;
#pragma unroll
      for (int kh = 0; kh < 3; ++kh) {
        int ih = oh * 2 - 1 + kh;
        if (ih < 0 || ih >= HIN_) continue;
#pragma unroll
        for (int kw = 0; kw < 3; ++kw) {
          int iw = ow * 2 - 1 + kw;
          if (iw < 0 || iw >= WIN_) continue;
          acc += xp[ih * WIN_ + iw] * sw[ci * 9 + kh * 3 + kw];
        }
      }
    }
    acc = acc > 0.f ? acc : 0.f;
    vals[p] = acc;
    ss += acc * acc;
  }
  sred[tid] = ss;
  __syncthreads();
  for (int s = 128; s > 0; s >>= 1) {
    if (tid < s) sred[tid] += sred[tid + s];
    __syncthreads();
  }
  float inv = 1.f / (sqrtf(sred[0]) + 1e-6f);
#pragma unroll
  for (int p = 0; p < 4; ++p)
    af[(size_t)bc * N_ + tid + p * 256] = vals[p] * inv;
}

// ---------------------------------------------------------------------------
// Kernel 2: delta = (1e-3/C) * Af^T Af  via V_WMMA_F32_16X16X4_F32
// Af treated as [R_=2048, N_=1024]; each wave owns one 16x16 delta tile.
// 512 blocks * 8 waves = 4096 tiles = 64x64 tile grid.
// ---------------------------------------------------------------------------
__global__ void gram_kernel(const float* __restrict__ af,
                            float* __restrict__ delta) {
  int wave = blockIdx.x * 8 + (threadIdx.x >> 5);
  int lane = threadIdx.x & 31;
  int half = lane >> 4;      // K sub-pair selector per ISA A/B layout
  int l    = lane & 15;
  int i0 = (wave >> 6) * 16;
  int j0 = (wave & 63) * 16;
  v8f acc = {};
  const float* pa = af + i0 + l;
  const float* pb = af + j0 + l;
  for (int k = 0; k < R_; k += 4) {
    int r0 = k + 2 * half;
    v2f a, bm;
    a.x  = pa[(size_t)r0 * N_];
    a.y  = pa[(size_t)(r0 + 1) * N_];
    bm.x = pb[(size_t)r0 * N_];
    bm.y = pb[(size_t)(r0 + 1) * N_];
    acc = __builtin_amdgcn_wmma_f32_16x16x4_f32(false, a, false, bm,
                                                (short)0, acc, false, false);
  }
  const float scale = 1e-3f / (float)C_;
#pragma unroll
  for (int r = 0; r < 8; ++r) {
    int mi = i0 + r + 8 * half;
    delta[(size_t)mi * N_ + j0 + l] = acc[r] * scale;
  }
}

// ---------------------------------------------------------------------------
// Kernel 3: the 256 MB stream.  For each channel c:
//   lat[b, j] = sum_i af[b,c,i] * clip(lw[c,i,j] + delta[i,j], -1, 1)
// One block = (channel, 128-column slab); 8 waves each own 16 columns.
// Each wave loads one clipped 4x16 B-tile and feeds it to TWO wmmas
// (batch rows 0..15 and 16..31) so lw is read exactly once from HBM.
// ---------------------------------------------------------------------------
__global__ void lateral_kernel(const float* __restrict__ af,
                               const float* __restrict__ lw,
                               const float* __restrict__ delta,
                               float* __restrict__ lat) {
  int c  = blockIdx.x >> 3;
  int jb = blockIdx.x & 7;
  int wi   = threadIdx.x >> 5;
  int lane = threadIdx.x & 31;
  int half = lane >> 4;
  int l    = lane & 15;
  int j0 = jb * 128 + wi * 16;
  v8f c0 = {}, c1 = {};
  const float* lwc = lw + (size_t)c * N_ * N_ + j0 + l;
  const float* dl  = delta + j0 + l;
  const float* a0p = af + ((size_t)l * C_ + c) * N_;        // batch rows 0..15
  const float* a1p = af + ((size_t)(l + 16) * C_ + c) * N_; // batch rows 16..31
  for (int k = 0; k < N_; k += 4) {
    int r0 = k + 2 * half;
    // prefetch the lw rows 8 ahead in the stream (global_prefetch_b8)
    __builtin_prefetch(lwc + (size_t)(r0 + 8) * N_, 0, 1);
    v2f bm, a0, a1;
    float w0 = lwc[(size_t)r0 * N_]       + dl[(size_t)r0 * N_];
    float w1 = lwc[(size_t)(r0 + 1) * N_] + dl[(size_t)(r0 + 1) * N_];
    bm.x = fminf(fmaxf(w0, -1.f), 1.f);
    bm.y = fminf(fmaxf(w1, -1.f), 1.f);
    a0.x = a0p[r0]; a0.y = a0p[r0 + 1];
    a1.x = a1p[r0]; a1.y = a1p[r0 + 1];
    c0 = __builtin_amdgcn_wmma_f32_16x16x4_f32(false, a0, false, bm,
                                               (short)0, c0, false, false);
    c1 = __builtin_amdgcn_wmma_f32_16x16x4_f32(false, a1, false, bm,
                                               (short)0, c1, false, false);
  }
#pragma unroll
  for (int r = 0; r < 8; ++r) {
    int m = r + 8 * half;
    lat[((size_t)m * C_ + c) * N_ + j0 + l]        = c0[r];
    lat[((size_t)(m + 16) * C_ + c) * N_ + j0 + l] = c1[r];
  }
}

// ---------------------------------------------------------------------------
// Kernel 4: out = af + lateral - mean_spatial(lateral)
// ---------------------------------------------------------------------------
__global__ void finalize_kernel(const float* __restrict__ af,
                                const float* __restrict__ lat,
                                float* __restrict__ out) {
  int bc = blockIdx.x;
  __shared__ float sred[256];
  int tid = threadIdx.x;
  float lv[4];
  float s = 0.f;
#pragma unroll
  for (int p = 0; p < 4; ++p) {
    lv[p] = lat[(size_t)bc * N_ + tid + p * 256];
    s += lv[p];
  }
  sred[tid] = s;
  __syncthreads();
  for (int st = 128; st > 0; st >>= 1) {
    if (tid < st) sred[tid] += sred[tid + st];
    __syncthreads();
  }
  float mean = sred[0] * (1.f / (float)N_);
#pragma unroll
  for (int p = 0; p < 4; ++p) {
    size_t idx = (size_t)bc * N_ + tid + p * 256;
    out[idx] = af[idx] + lv[p] - mean;
  }
}

extern "C" void kernel_launch(void* const* d_in, const int* in_sizes, int n_in,
                              void* d_out, int out_size, void* d_ws, size_t ws_size,
                              hipStream_t stream) {
  const float* x      = (const float*)d_in[0];
  const float* conv_w = (const float*)d_in[1];
  const float* conv_b = (const float*)d_in[2];
  const float* lw     = (const float*)d_in[3];
  float* out = (float*)d_out;

  float* af    = (float*)d_ws;                 // [2048, 1024]  8 MB
  float* delta = af + (size_t)R_ * N_;         // [1024, 1024]  4 MB
  float* lat   = delta + (size_t)N_ * N_;      // [2048, 1024]  8 MB

  conv_norm_kernel<<<R_, 256, 0, stream>>>(x, conv_w, conv_b, af);
  gram_kernel<<<512, 256, 0, stream>>>(af, delta);
  lateral_kernel<<<C_ * 8, 256, 0, stream>>>(af, lw, delta, lat);
  finalize_kernel<<<R_, 256, 0, stream>>>(af, lat, out);
}